// _IPEXAttentionRef_34823594836587
// MI455X (gfx1250) — compile-verified
//
#include <hip/hip_runtime.h>
#include <math.h>

typedef __attribute__((ext_vector_type(2))) float v2f;
typedef __attribute__((ext_vector_type(8))) float v8f;

#define DMODEL 4096
#define BATCH  8
#define HEADS  16
#define DH     256
#define PAST   4096
#define NKEYS  (PAST + 1)
#define SPLITS 8
#define LDIM   36   /* LDS row stride in floats: 16B aligned, bank-conflict free */
#define KSTAGE 32
#define GWAVES 4
#define KRANGE (DMODEL / GWAVES)   /* 1024 per wave */

// -------- CDNA5 async memory->LDS helpers (ASYNCcnt-tracked, no VGPR round trip)
__device__ __forceinline__ void async_copy_b128(unsigned int lds_byte_addr, const void* gsrc)
{
    // GLOBAL_LOAD_ASYNC_TO_LDS_B128: vdst = LDS byte address, vaddr = 64-bit global addr
    asm volatile("global_load_async_to_lds_b128 %0, %1, off"
                 :: "v"(lds_byte_addr), "v"(gsrc) : "memory");
}
#define WAIT_ASYNCCNT(n) asm volatile("s_wait_asynccnt %0" :: "i"(n) : "memory")
#define WAIT_DSCNT0()    asm volatile("s_wait_dscnt 0x0" ::: "memory")

// ---------------------------------------------------------------------------
// Y[8 x 4096] = X[8 x 4096] @ W^T, W row-major [4096 x 4096].
// One block = one 16-column tile of Y; 4 waves split K (1024 each).
// Double-buffered per-wave LDS tiles filled by global_load_async_to_lds_b128;
// D = A(16x4, batch rows padded to 16) x B(4x16) via v_wmma_f32_16x16x4_f32.
// Wave-private tiles => counter-only sync (no barriers); last stage peeled so
// the steady-state loop body is branch-free: issue -> wait 6 -> 8xWMMA -> wait ds.
// ---------------------------------------------------------------------------
__global__ __launch_bounds__(128) void gemm_wmma(
    const float* __restrict__ X,
    const float* __restrict__ W0, const float* __restrict__ W1, const float* __restrict__ W2,
    float* __restrict__ Y0, float* __restrict__ Y1, float* __restrict__ Y2)
{
    __shared__ float wt[GWAVES][2][16 * LDIM];
    __shared__ float xt[GWAVES][2][16 * LDIM];
    __shared__ float red[GWAVES][8][16];

    const int w  = threadIdx.x >> 5;   // wave id 0..3 (k-split)
    const int ln = threadIdx.x & 31;   // lane 0..31

    const float* W = (blockIdx.y == 0) ? W0 : ((blockIdx.y == 1) ? W1 : W2);
    float*       Y = (blockIdx.y == 0) ? Y0 : ((blockIdx.y == 1) ? Y1 : Y2);

    const int n0 = blockIdx.x * 16;

    // zero the pad rows (batch rows 8..15) of both X buffers once
    for (int buf = 0; buf < 2; ++buf)
        for (int i = ln; i < 8 * LDIM; i += 32)
            xt[w][buf][8 * LDIM + i] = 0.0f;

    const int kbeg = w * KRANGE;

    // issue one stage: 4 async b128 for the W tile (16x32) + 2 for the X tile (8x32)
    auto issue = [&](int k0, int buf) {
        unsigned int wdst = (unsigned int)(uintptr_t)(&wt[w][buf][0]);
        unsigned int xdst = (unsigned int)(uintptr_t)(&xt[w][buf][0]);
        #pragma unroll
        for (int i = 0; i < 4; ++i) {
            int t = ln + 32 * i;
            int row = t >> 3;
            int c4  = (t & 7) << 2;
            async_copy_b128(wdst + (unsigned int)((row * LDIM + c4) * 4),
                            W + (size_t)(n0 + row) * DMODEL + k0 + c4);
        }
        #pragma unroll
        for (int i = 0; i < 2; ++i) {
            int t = ln + 32 * i;
            int row = t >> 3;
            int c4  = (t & 7) << 2;
            async_copy_b128(xdst + (unsigned int)((row * LDIM + c4) * 4),
                            X + (size_t)row * DMODEL + k0 + c4);
        }
    };

    v8f c = {};
    const int r    = ln & 15;          // M row (A) / N col (B)
    const int koff = (ln >> 4) << 1;   // lanes 16-31 carry K+2

    auto mma_stage = [&](int buf) {
        const float* wtile = &wt[w][buf][0];
        const float* xtile = &xt[w][buf][0];
        #pragma unroll
        for (int kk = 0; kk < KSTAGE; kk += 4) {
            v2f a, b;
            a.x = xtile[r * LDIM + kk + koff];
            a.y = xtile[r * LDIM + kk + koff + 1];
            b.x = wtile[r * LDIM + kk + koff];
            b.y = wtile[r * LDIM + kk + koff + 1];
            c = __builtin_amdgcn_wmma_f32_16x16x4_f32(
                    false, a, false, b, (short)0, c, false, false);
        }
    };

    issue(kbeg, 0);
    int parity = 0;

    // steady state: prefetch next stage, wait for current (in-order completion),
    // compute, then ensure fragment reads done before the buffer is re-filled.
    for (int k0 = kbeg; k0 < kbeg + KRANGE - KSTAGE; k0 += KSTAGE) {
        issue(k0 + KSTAGE, parity ^ 1);
        WAIT_ASYNCCNT(6);
        mma_stage(parity);
        WAIT_DSCNT0();
        parity ^= 1;
    }
    // final (peeled) stage
    WAIT_ASYNCCNT(0);
    mma_stage(parity);

    // D layout: lanes 0-15 hold N=lane, M = vgpr j (batch rows 0..7)
    if (ln < 16) {
        #pragma unroll
        for (int j = 0; j < 8; ++j) red[w][j][ln] = c[j];
    }
    __syncthreads();

    {
        int j = threadIdx.x >> 4;   // 0..7
        int n = threadIdx.x & 15;
        float sum = 0.0f;
        #pragma unroll
        for (int i = 0; i < GWAVES; ++i) sum += red[i][j][n];
        Y[(size_t)j * DMODEL + n0 + n] = sum;
    }
}

// ---------------------------------------------------------------------------
// GPT-J rotate_every_two RoPE on first 64 dims of q and k.
// ---------------------------------------------------------------------------
__global__ __launch_bounds__(32) void rope_kernel(float* __restrict__ q,
                                                  float* __restrict__ k,
                                                  const int* __restrict__ pos)
{
    const int b = blockIdx.x >> 4;
    const int h = blockIdx.x & 15;
    const int i = threadIdx.x;                 // pair index 0..31
    const float p = (float)pos[b];
    const float freq = expf(-((float)(2 * i) / 64.0f) * 9.210340371976184f); // ln(10000)
    const float ang = p * freq;
    const float s = sinf(ang);
    const float cc = cosf(ang);
    const size_t base = (size_t)b * DMODEL + h * DH + 2 * i;

    float q0 = q[base], q1 = q[base + 1];
    q[base]     = q0 * cc - q1 * s;
    q[base + 1] = q1 * cc + q0 * s;
    float k0 = k[base], k1 = k[base + 1];
    k[base]     = k0 * cc - k1 * s;
    k[base + 1] = k1 * cc + k0 * s;
}

// ---------------------------------------------------------------------------
// Flash-decoding partial: grid (SPLITS, B*H), 8 waves per block.
// Each wave streams keys (whole 1KB K row per key, float4 coalesced),
// butterfly-reduces the dot, runs online softmax, accumulates V.
// ---------------------------------------------------------------------------
__global__ __launch_bounds__(256) void attn_partial(
    const float* __restrict__ q, const float* __restrict__ knew, const float* __restrict__ vnew,
    const float* __restrict__ pk, const float* __restrict__ pv,
    const float* __restrict__ mask, float* __restrict__ part)
{
    const int s  = blockIdx.x;          // split
    const int bh = blockIdx.y;          // b*16 + h
    const int b  = bh >> 4;
    const int h  = bh & 15;
    const int w  = threadIdx.x >> 5;
    const int ln = threadIdx.x & 31;

    const int chunk = (NKEYS + SPLITS - 1) / SPLITS;   // 513
    const int start = s * chunk;
    const int end   = (start + chunk < NKEYS) ? (start + chunk) : NKEYS;

    const float* qp = q + (size_t)b * DMODEL + h * DH;
    const float4 q0 = *(const float4*)(qp + ln * 4);
    const float4 q1 = *(const float4*)(qp + 128 + ln * 4);

    const float scale = 0.0625f;  // 1/sqrt(256)
    float m = -3.0e38f, l = 0.0f;
    float4 a0 = {0, 0, 0, 0}, a1 = {0, 0, 0, 0};

    const float* kbase = pk + (size_t)bh * PAST * DH;
    const float* vbase = pv + (size_t)bh * PAST * DH;
    const float* kn = knew + (size_t)b * DMODEL + h * DH;
    const float* vn = vnew + (size_t)b * DMODEL + h * DH;
    const float* mrow = mask + (size_t)b * NKEYS;

    for (int key = start + w; key < end; key += 8) {
        const float* kp = (key < PAST) ? (kbase + (size_t)key * DH) : kn;
        const float* vp = (key < PAST) ? (vbase + (size_t)key * DH) : vn;
        float4 kk0 = *(const float4*)(kp + ln * 4);
        float4 kk1 = *(const float4*)(kp + 128 + ln * 4);
        float d = q0.x * kk0.x + q0.y * kk0.y + q0.z * kk0.z + q0.w * kk0.w
                + q1.x * kk1.x + q1.y * kk1.y + q1.z * kk1.z + q1.w * kk1.w;
        #pragma unroll
        for (int off = 16; off > 0; off >>= 1) d += __shfl_xor(d, off, 32);

        float sc = d * scale + mrow[key];
        float mn = fmaxf(m, sc);
        float corr = __expf(m - mn);
        float p    = __expf(sc - mn);
        l = l * corr + p;

        float4 vv0 = *(const float4*)(vp + ln * 4);
        float4 vv1 = *(const float4*)(vp + 128 + ln * 4);
        a0.x = a0.x * corr + p * vv0.x;  a0.y = a0.y * corr + p * vv0.y;
        a0.z = a0.z * corr + p * vv0.z;  a0.w = a0.w * corr + p * vv0.w;
        a1.x = a1.x * corr + p * vv1.x;  a1.y = a1.y * corr + p * vv1.y;
        a1.z = a1.z * corr + p * vv1.z;  a1.w = a1.w * corr + p * vv1.w;
        m = mn;
    }

    __shared__ float sm[8], sl[8];
    __shared__ float sacc[8][256];
    if (ln == 0) { sm[w] = m; sl[w] = l; }
    *(float4*)&sacc[w][ln * 4]       = a0;
    *(float4*)&sacc[w][128 + ln * 4] = a1;
    __syncthreads();

    if (w == 0) {
        float M = sm[0];
        #pragma unroll
        for (int i = 1; i < 8; ++i) M = fmaxf(M, sm[i]);
        float L = 0.0f;
        float4 t0 = {0, 0, 0, 0}, t1 = {0, 0, 0, 0};
        #pragma unroll
        for (int i = 0; i < 8; ++i) {
            float e = __expf(sm[i] - M);
            L += e * sl[i];
            float4 c0 = *(float4*)&sacc[i][ln * 4];
            float4 c1 = *(float4*)&sacc[i][128 + ln * 4];
            t0.x += e * c0.x; t0.y += e * c0.y; t0.z += e * c0.z; t0.w += e * c0.w;
            t1.x += e * c1.x; t1.y += e * c1.y; t1.z += e * c1.z; t1.w += e * c1.w;
        }
        float* pb = part + ((size_t)bh * SPLITS + s) * 260;
        if (ln == 0) { pb[0] = M; pb[1] = L; }
        *(float4*)(pb + 4 + ln * 4)       = t0;
        *(float4*)(pb + 4 + 128 + ln * 4) = t1;
    }
}

// ---------------------------------------------------------------------------
// Combine the SPLITS partials per (b,h) -> ctx[b][h*256 + d]
// ---------------------------------------------------------------------------
__global__ __launch_bounds__(32) void attn_reduce(const float* __restrict__ part,
                                                  float* __restrict__ ctx)
{
    const int bh = blockIdx.x;
    const int ln = threadIdx.x;

    float M = -3.0e38f;
    for (int s = 0; s < SPLITS; ++s)
        M = fmaxf(M, part[((size_t)bh * SPLITS + s) * 260]);

    float L = 0.0f;
    float4 t0 = {0, 0, 0, 0}, t1 = {0, 0, 0, 0};
    for (int s = 0; s < SPLITS; ++s) {
        const float* pb = part + ((size_t)bh * SPLITS + s) * 260;
        float e = __expf(pb[0] - M);
        L += e * pb[1];
        float4 c0 = *(const float4*)(pb + 4 + ln * 4);
        float4 c1 = *(const float4*)(pb + 4 + 128 + ln * 4);
        t0.x += e * c0.x; t0.y += e * c0.y; t0.z += e * c0.z; t0.w += e * c0.w;
        t1.x += e * c1.x; t1.y += e * c1.y; t1.z += e * c1.z; t1.w += e * c1.w;
    }
    const float inv = 1.0f / L;
    const int b = bh >> 4, h = bh & 15;
    float* cp = ctx + (size_t)b * DMODEL + h * DH;
    float4 o0 = {t0.x * inv, t0.y * inv, t0.z * inv, t0.w * inv};
    float4 o1 = {t1.x * inv, t1.y * inv, t1.z * inv, t1.w * inv};
    *(float4*)(cp + ln * 4)       = o0;
    *(float4*)(cp + 128 + ln * 4) = o1;
}

// ---------------------------------------------------------------------------
extern "C" void kernel_launch(void* const* d_in, const int* in_sizes, int n_in,
                              void* d_out, int out_size, void* d_ws, size_t ws_size,
                              hipStream_t stream)
{
    const float* hs   = (const float*)d_in[0];   // [8,1,4096]
    const float* pk   = (const float*)d_in[1];   // [8,16,4096,256]
    const float* pv   = (const float*)d_in[2];   // [8,16,4096,256]
    const float* mask = (const float*)d_in[3];   // [8,1,1,4097]
    const int*   pos  = (const int*)d_in[4];     // [8,1] (jax default int32)
    const float* Wq   = (const float*)d_in[5];
    const float* Wk   = (const float*)d_in[6];
    const float* Wv   = (const float*)d_in[7];
    const float* Wo   = (const float*)d_in[8];
    float* out = (float*)d_out;                  // [8,1,4096] fp32

    float* ws   = (float*)d_ws;
    float* q    = ws;                 // 32768 floats
    float* k    = ws + 32768;         // 32768
    float* v    = ws + 65536;         // 32768
    float* ctx  = ws + 98304;         // 32768
    float* part = ws + 131072;        // 128 * 8 * 260 = 266240 floats  (~1.6 MB total)

    // 1) fused QKV projection (WMMA f32 16x16x4, async-LDS double-buffered)
    gemm_wmma<<<dim3(256, 3), 128, 0, stream>>>(hs, Wq, Wk, Wv, q, k, v);

    // 2) RoPE on q and k (first 64 dims per head)
    rope_kernel<<<BATCH * HEADS, 32, 0, stream>>>(q, k, pos);

    // 3) flash-decoding attention over 4096 past + 1 new key
    attn_partial<<<dim3(SPLITS, BATCH * HEADS), 256, 0, stream>>>(q, k, v, pk, pv, mask, part);
    attn_reduce<<<BATCH * HEADS, 32, 0, stream>>>(part, ctx);

    // 4) output projection
    gemm_wmma<<<dim3(256, 1), 128, 0, stream>>>(ctx, Wo, Wo, Wo, out, out, out);
}